// EnhanceDiversityFeatureExtracition_16303695856051
// MI455X (gfx1250) — compile-verified
//
#include <hip/hip_runtime.h>

typedef __attribute__((ext_vector_type(2))) float v2f;
typedef __attribute__((ext_vector_type(8))) float v8f;

#define F_DIM        128
#define SLICE_STRIDE 384          // 128 filters * 3 taps, contiguous per (o,i) slice
#define CHUNK        32           // slices staged per LDS chunk
#define CHUNK_K      (CHUNK * 3)  // 96 K-values per chunk
#define RS           (CHUNK_K + 1)// padded LDS row stride (97: odd -> conflict-free)
#define N_SLICES     65536        // 256*256
#define N_CHUNKS     (N_SLICES / CHUNK)
#define GRAM_OFF     64           // float offset of Gram matrix inside d_ws
#define ALPHA_C      0.0005f
#define TAU_C        0.2f

// ---------------------------------------------------------------- zero scratch
__global__ void zero_ws(float* ws, int n) {
    for (int i = blockIdx.x * blockDim.x + threadIdx.x; i < n;
         i += gridDim.x * blockDim.x)
        ws[i] = 0.0f;
}

// ---------------------------------------------------------------- MSE reduce
__global__ __launch_bounds__(256)
void mse_partial(const float4* __restrict__ a, const float4* __restrict__ b,
                 float* __restrict__ accum, int n4) {
    __shared__ float red[256];
    float s = 0.0f;
    for (int i = blockIdx.x * blockDim.x + threadIdx.x; i < n4;
         i += gridDim.x * blockDim.x) {
        float4 x = a[i], y = b[i];
        float d0 = x.x - y.x, d1 = x.y - y.y, d2 = x.z - y.z, d3 = x.w - y.w;
        s += d0 * d0 + d1 * d1 + d2 * d2 + d3 * d3;
    }
    red[threadIdx.x] = s;
    __syncthreads();
    for (int off = 128; off > 0; off >>= 1) {
        if ((int)threadIdx.x < off) red[threadIdx.x] += red[threadIdx.x + off];
        __syncthreads();
    }
    if (threadIdx.x == 0) atomicAdd(accum, red[0]);
}

// ---------------------------------------------------------------- Gram (SYRK) via WMMA f32 16x16x4
// gram[i][j] = sum_d V[i][d]*V[j][d],  V[f][(o,i,k)] = conv_w[o][i][f][k]
__global__ __launch_bounds__(256)
void gram_wmma(const float* __restrict__ w, float* __restrict__ gram) {
    __shared__ float lds[F_DIM * RS];   // 128 * 97 * 4B = 49664 B

    const int tid  = threadIdx.x;
    const int wave = tid >> 5;          // 0..7 : A tile row (ti)
    const int lane = tid & 31;
    const int half = lane >> 4;         // 0: K={0,1}, 1: K={2,3}
    const int m    = lane & 15;         // matrix row (A) / col (B/C)

    v8f acc[8] = {};                    // 8 column tiles, f32 16x16 each

    for (int c = blockIdx.x; c < N_CHUNKS; c += gridDim.x) {
        // ---- stage chunk: global [slice][f][k] -> LDS [f][slice*3+k] (padded)
        const float* src = w + (size_t)c * (CHUNK * SLICE_STRIDE);
        for (int idx = tid; idx < CHUNK * SLICE_STRIDE; idx += 256) {
            int s_local = idx / SLICE_STRIDE;
            int rem     = idx - s_local * SLICE_STRIDE;
            int f       = rem / 3;
            int r       = rem - f * 3;
            lds[f * RS + s_local * 3 + r] = src[idx];
        }
        __syncthreads();

        // ---- accumulate A*A^T over this chunk's 96 K-values
        // A frag: lane<16 -> (M=m, K=kk+{0,1}); lane>=16 -> (M=m, K=kk+{2,3})
        // B frag mirrors A (B[k][n] = V[tile_n*16+n][k])  => D = Gram tile
        const float* arow = &lds[(wave * 16 + m) * RS + 2 * half];
        for (int kk = 0; kk < CHUNK_K; kk += 4) {
            v2f a;
            a.x = arow[kk];
            a.y = arow[kk + 1];
#pragma unroll
            for (int tj = 0; tj < 8; ++tj) {
                const float* brow = &lds[(tj * 16 + m) * RS + 2 * half];
                v2f b;
                b.x = brow[kk];
                b.y = brow[kk + 1];
                acc[tj] = __builtin_amdgcn_wmma_f32_16x16x4_f32(
                    /*neg_a=*/false, a, /*neg_b=*/false, b,
                    /*c_mod=*/(short)0, acc[tj],
                    /*reuse_a=*/false, /*reuse_b=*/false);
            }
        }
        __syncthreads();
    }

    // ---- C/D layout: VGPR r -> M=r (lanes 0-15) or M=r+8 (lanes 16-31); N=m
#pragma unroll
    for (int tj = 0; tj < 8; ++tj) {
#pragma unroll
        for (int r = 0; r < 8; ++r) {
            int row = wave * 16 + r + 8 * half;
            int col = tj * 16 + m;
            atomicAdd(&gram[row * F_DIM + col], acc[tj][r]);
        }
    }
}

// ---------------------------------------------------------------- finalize
__global__ __launch_bounds__(256)
void finalize_loss(const float* __restrict__ gram, const float* __restrict__ mse_acc,
                   float* __restrict__ out, float inv_n) {
    __shared__ float inv_norm[F_DIM];
    __shared__ float red[256];
    const int tid = threadIdx.x;

    if (tid < F_DIM) inv_norm[tid] = rsqrtf(gram[tid * F_DIM + tid]);
    __syncthreads();

    float s = 0.0f;
    for (int idx = tid; idx < F_DIM * F_DIM; idx += 256) {
        int i = idx >> 7, j = idx & 127;
        if (i != j) {
            float cg = gram[idx] * inv_norm[i] * inv_norm[j];
            if (cg > TAU_C && cg <= 1.0f) s += cg;
        }
    }
    red[tid] = s;
    __syncthreads();
    for (int off = 128; off > 0; off >>= 1) {
        if (tid < off) red[tid] += red[tid + off];
        __syncthreads();
    }
    if (tid == 0) out[0] = mse_acc[0] * inv_n + ALPHA_C * red[0];
}

// ---------------------------------------------------------------- launch
extern "C" void kernel_launch(void* const* d_in, const int* in_sizes, int n_in,
                              void* d_out, int out_size, void* d_ws, size_t ws_size,
                              hipStream_t stream) {
    (void)n_in; (void)out_size; (void)ws_size;
    const float* output = (const float*)d_in[0];
    const float* target = (const float*)d_in[1];
    const float* conv_w = (const float*)d_in[2];

    float* ws      = (float*)d_ws;
    float* mse_acc = ws;             // ws[0]
    float* gram    = ws + GRAM_OFF;  // ws[64 .. 64+16384)

    const int n  = in_sizes[0];      // 8192*1000
    const int n4 = n / 4;

    zero_ws<<<64, 256, 0, stream>>>(ws, GRAM_OFF + F_DIM * F_DIM);
    mse_partial<<<1024, 256, 0, stream>>>((const float4*)output,
                                          (const float4*)target, mse_acc, n4);
    gram_wmma<<<256, 256, 0, stream>>>(conv_w, gram);
    finalize_loss<<<1, 256, 0, stream>>>(gram, mse_acc, (float*)d_out,
                                         1.0f / (float)n);
}